// EdgeConv_41351945126431
// MI455X (gfx1250) — compile-verified
//
#include <hip/hip_runtime.h>
#include <cstdint>

#define KNN   20
#define Bsz   8
#define Cdim  64
#define Npts  4096
#define OUTC  64

typedef __attribute__((ext_vector_type(2))) float v2f;
typedef __attribute__((ext_vector_type(8))) float v8f;

// ---------------------------------------------------------------- sq norms
__global__ __launch_bounds__(256) void sqnorm_kernel(const float* __restrict__ x,
                                                     float* __restrict__ sq) {
  int gid = blockIdx.x * 256 + threadIdx.x;     // b*N + n
  int b = gid >> 12;
  int n = gid & (Npts - 1);
  const float* xb = x + (size_t)b * Cdim * Npts + n;
  float s = 0.f;
  #pragma unroll
  for (int c = 0; c < Cdim; ++c) { float v = xb[(size_t)c * Npts]; s = fmaf(v, v, s); }
  sq[gid] = s;
}

// ---------------------------------------------------------------- fused Gram + top-K
__global__ __launch_bounds__(256) void knn_kernel(const float* __restrict__ x,
                                                  const float* __restrict__ sq,
                                                  int* __restrict__ idxout) {
  __shared__ float xpan[2][Cdim * 16];  // double-buffered column panel [k][n]
  __shared__ float sqc[2][16];
  __shared__ float gt[8][16 * 17];      // per-wave Gram tile (padded)
  __shared__ float md[8][16][KNN];      // half-wave merge buffers
  __shared__ int   mi[8][16][KNN];

  const int b    = blockIdx.y;
  const int lane = threadIdx.x & 31;
  const int wave = threadIdx.x >> 5;
  const int rl   = lane & 15;           // row within tile
  const int kh   = lane >> 4;           // K-half (f32 WMMA A layout)
  const int row  = blockIdx.x * 128 + wave * 16 + rl;
  const float* xb = x + (size_t)b * Cdim * Npts;

  // Async global->LDS staging of one 64x16 fp32 column panel (16B per lane).
  const int pk  = threadIdx.x >> 2;        // k row  (0..63)
  const int pc4 = (threadIdx.x & 3) * 4;   // 4-col group
  auto issue_panel = [&](int buf, int ct) {
    const int col0 = ct * 16;
    uint32_t lds_addr = (uint32_t)(uintptr_t)&xpan[buf][pk * 16 + pc4];
    const float* gp = xb + (size_t)pk * Npts + col0 + pc4;
    asm volatile("global_load_async_to_lds_b128 %0, %1, off"
                 :: "v"(lds_addr), "v"(gp) : "memory");
    if (threadIdx.x < 16) sqc[buf][threadIdx.x] = sq[b * Npts + col0 + threadIdx.x];
  };

  // Preload full A row-panel (16 rows x 64 K) into registers: reused for 256 tiles.
  v2f apan[16];
  #pragma unroll
  for (int s = 0; s < 16; ++s) {
    int k0 = 4 * s + 2 * kh;
    v2f a;
    a.x = xb[(size_t)(k0 + 0) * Npts + row];
    a.y = xb[(size_t)(k0 + 1) * Npts + row];
    apan[s] = a;
  }
  const float sqr = sq[b * Npts + row];

  float dl[KNN]; int il[KNN];
  #pragma unroll
  for (int j = 0; j < KNN; ++j) { dl[j] = 1e30f; il[j] = 0; }

  issue_panel(0, 0);

  for (int ct = 0; ct < Npts / 16; ++ct) {
    const int buf  = ct & 1;
    const int col0 = ct * 16;
    // Retire this wave's outstanding async copy, then make panel visible block-wide.
    asm volatile("s_wait_asynccnt 0" ::: "memory");
    __syncthreads();
    // Prefetch next tile's panel into the other buffer (overlaps with compute).
    if (ct + 1 < Npts / 16) issue_panel(buf ^ 1, ct + 1);

    const float* xp = &xpan[buf][0];
    v8f acc = {};
    #pragma unroll
    for (int s = 0; s < 16; ++s) {
      int kb = 4 * s + 2 * kh;
      v2f bf;
      bf.x = xp[(kb + 0) * 16 + rl];
      bf.y = xp[(kb + 1) * 16 + rl];
      acc = __builtin_amdgcn_wmma_f32_16x16x4_f32(false, apan[s], false, bf,
                                                  (short)0, acc, false, false);
    }

    // Transpose C layout (row = v + 8*kh, col = rl) to row-major via LDS.
    float* g = &gt[wave][0];
    #pragma unroll
    for (int v = 0; v < 8; ++v) g[(v + 8 * kh) * 17 + rl] = acc[v];
    __syncthreads();

    // Each half-wave scans 8 columns for its row; register-resident sorted top-K.
    #pragma unroll 1
    for (int cc = 0; cc < 8; ++cc) {
      int c = kh * 8 + cc;
      float d = sqr + sqc[buf][c] - 2.f * g[rl * 17 + c];
      if (d < dl[KNN - 1]) {
        float dd = d; int ii = col0 + c;
        #pragma unroll
        for (int j = 0; j < KNN; ++j) {
          if (dd < dl[j]) {
            float td = dl[j]; dl[j] = dd; dd = td;
            int   ti = il[j]; il[j] = ii; ii = ti;
          }
        }
      }
    }
  }

  // Merge the two half-wave lists.
  __syncthreads();
  if (kh == 1) {
    #pragma unroll
    for (int j = 0; j < KNN; ++j) { md[wave][rl][j] = dl[j]; mi[wave][rl][j] = il[j]; }
  }
  __syncthreads();
  if (kh == 0) {
    #pragma unroll 1
    for (int j = 0; j < KNN; ++j) {
      float d = md[wave][rl][j]; int ic = mi[wave][rl][j];
      if (d < dl[KNN - 1]) {
        float dd = d; int ii = ic;
        #pragma unroll
        for (int q = 0; q < KNN; ++q) {
          if (dd < dl[q]) {
            float td = dl[q]; dl[q] = dd; dd = td;
            int   ti = il[q]; il[q] = ii; ii = ti;
          }
        }
      }
    }
    int* op = idxout + ((size_t)b * Npts + row) * KNN;
    #pragma unroll
    for (int j = 0; j < KNN; ++j) op[j] = il[j];
  }
}

// ---------------------------------------------------------------- h = [W1-W2; W2] @ x
__global__ __launch_bounds__(256) void feat_gemm_kernel(const float* __restrict__ x,
                                                        const float* __restrict__ W,
                                                        float* __restrict__ h) {
  const int b    = blockIdx.y;
  const int lane = threadIdx.x & 31;
  const int wave = threadIdx.x >> 5;
  const int rl   = lane & 15;
  const int kh   = lane >> 4;
  const int m    = wave * 16 + rl;      // 0..127 combined channel
  const float* xb = x + (size_t)b * Cdim * Npts;

  v2f apan[16];
  #pragma unroll
  for (int s = 0; s < 16; ++s) {
    int k0 = 4 * s + 2 * kh;
    v2f a;
    if (m < OUTC) {
      a.x = W[m * 128 + k0]     - W[m * 128 + 64 + k0];
      a.y = W[m * 128 + k0 + 1] - W[m * 128 + 64 + k0 + 1];
    } else {
      a.x = W[(m - OUTC) * 128 + 64 + k0];
      a.y = W[(m - OUTC) * 128 + 64 + k0 + 1];
    }
    apan[s] = a;
  }

  for (int t = 0; t < 8; ++t) {
    int n0 = blockIdx.x * 128 + t * 16;
    v8f acc = {};
    #pragma unroll
    for (int s = 0; s < 16; ++s) {
      int kb = 4 * s + 2 * kh;
      v2f bf;
      bf.x = xb[(size_t)(kb + 0) * Npts + n0 + rl];
      bf.y = xb[(size_t)(kb + 1) * Npts + n0 + rl];
      acc = __builtin_amdgcn_wmma_f32_16x16x4_f32(false, apan[s], false, bf,
                                                  (short)0, acc, false, false);
    }
    #pragma unroll
    for (int v = 0; v < 8; ++v) {
      int mm = wave * 16 + v + 8 * kh;
      h[((size_t)b * 128 + mm) * Npts + n0 + rl] = acc[v];
    }
  }
}

// ---------------------------------------------------------------- BN partial stats
__global__ __launch_bounds__(256) void bnstat_kernel(const float* __restrict__ h,
                                                     const int* __restrict__ idxb,
                                                     const float* __restrict__ bias,
                                                     float* __restrict__ partS,
                                                     float* __restrict__ partQ) {
  __shared__ float rs[256], rq[256];
  const int b = blockIdx.y;
  const int nbase = blockIdx.x * 256;
  const int o  = threadIdx.x & 63;
  const int ng = threadIdx.x >> 6;
  const float* h1 = h + ((size_t)b * 128 + o) * Npts;
  const float* h2 = h + ((size_t)b * 128 + OUTC + o) * Npts;
  const float bia = bias[o];
  float s = 0.f, q = 0.f;
  #pragma unroll 1
  for (int t = ng; t < 256; t += 4) {
    int n = nbase + t;
    const int* ip = idxb + ((size_t)b * Npts + n) * KNN;
    float base = h1[n] + bia;
    #pragma unroll
    for (int k = 0; k < KNN; ++k) {
      float v = base + h2[ip[k]];
      s += v; q = fmaf(v, v, q);
    }
  }
  rs[threadIdx.x] = s; rq[threadIdx.x] = q;
  __syncthreads();
  if (threadIdx.x < 64) {
    float S = rs[threadIdx.x] + rs[threadIdx.x + 64] + rs[threadIdx.x + 128] + rs[threadIdx.x + 192];
    float Q = rq[threadIdx.x] + rq[threadIdx.x + 64] + rq[threadIdx.x + 128] + rq[threadIdx.x + 192];
    int blk = b * 16 + blockIdx.x;
    partS[blk * 64 + threadIdx.x] = S;
    partQ[blk * 64 + threadIdx.x] = Q;
  }
}

// ---------------------------------------------------------------- BN finalize
__global__ __launch_bounds__(64) void bnfinal_kernel(const float* __restrict__ partS,
                                                     const float* __restrict__ partQ,
                                                     const float* __restrict__ gamma,
                                                     const float* __restrict__ beta,
                                                     float* __restrict__ scaleO,
                                                     float* __restrict__ shiftO) {
  int o = threadIdx.x;
  float S = 0.f, Q = 0.f;
  for (int blk = 0; blk < 128; ++blk) { S += partS[blk * 64 + o]; Q += partQ[blk * 64 + o]; }
  const float inv = 1.f / (float)((size_t)Bsz * Npts * KNN);
  float mean = S * inv;
  float var  = Q * inv - mean * mean;
  float sc   = gamma[o] * rsqrtf(var + 1e-5f);
  scaleO[o] = sc;
  shiftO[o] = beta[o] - mean * sc;
}

// ---------------------------------------------------------------- BN + LeakyReLU + max_k
__global__ __launch_bounds__(256) void edgeout_kernel(const float* __restrict__ h,
                                                      const int* __restrict__ idxb,
                                                      const float* __restrict__ bias,
                                                      const float* __restrict__ scaleO,
                                                      const float* __restrict__ shiftO,
                                                      float* __restrict__ out) {
  const int o = blockIdx.x;
  const int b = blockIdx.y;
  const float sc = scaleO[o], sh = shiftO[o], bia = bias[o];
  const float* h1 = h + ((size_t)b * 128 + o) * Npts;
  const float* h2 = h + ((size_t)b * 128 + OUTC + o) * Npts;
  float* op = out + ((size_t)b * OUTC + o) * Npts;
  #pragma unroll 1
  for (int n = threadIdx.x; n < Npts; n += 256) {
    const int* ip = idxb + ((size_t)b * Npts + n) * KNN;
    float t1 = fmaf(sc, h1[n] + bia, sh);
    float m = -3.4e38f;
    #pragma unroll
    for (int k = 0; k < KNN; ++k) {
      float v = fmaf(sc, h2[ip[k]], t1);
      v = (v >= 0.f) ? v : 0.2f * v;
      m = fmaxf(m, v);
    }
    op[n] = m;
  }
}

// ---------------------------------------------------------------- launcher
extern "C" void kernel_launch(void* const* d_in, const int* in_sizes, int n_in,
                              void* d_out, int out_size, void* d_ws, size_t ws_size,
                              hipStream_t stream) {
  (void)in_sizes; (void)n_in; (void)out_size; (void)ws_size;
  const float* x     = (const float*)d_in[0];
  const float* W     = (const float*)d_in[1];
  const float* bias  = (const float*)d_in[2];
  const float* gamma = (const float*)d_in[3];
  const float* beta  = (const float*)d_in[4];
  float* out = (float*)d_out;

  char* p = (char*)d_ws;
  float* sq   = (float*)p;  p += (size_t)Bsz * Npts * sizeof(float);
  int*  idxb  = (int*)p;    p += (size_t)Bsz * Npts * KNN * sizeof(int);
  float* h    = (float*)p;  p += (size_t)Bsz * 128 * Npts * sizeof(float);
  float* partS = (float*)p; p += (size_t)128 * 64 * sizeof(float);
  float* partQ = (float*)p; p += (size_t)128 * 64 * sizeof(float);
  float* scaleO = (float*)p; p += 64 * sizeof(float);
  float* shiftO = (float*)p; p += 64 * sizeof(float);

  sqnorm_kernel<<<dim3((Bsz * Npts) / 256), 256, 0, stream>>>(x, sq);
  knn_kernel<<<dim3(Npts / 128, Bsz), 256, 0, stream>>>(x, sq, idxb);
  feat_gemm_kernel<<<dim3(Npts / 128, Bsz), 256, 0, stream>>>(x, W, h);
  bnstat_kernel<<<dim3(16, Bsz), 256, 0, stream>>>(h, idxb, bias, partS, partQ);
  bnfinal_kernel<<<dim3(1), 64, 0, stream>>>(partS, partQ, gamma, beta, scaleO, shiftO);
  edgeout_kernel<<<dim3(OUTC, Bsz), 256, 0, stream>>>(h, idxb, bias, scaleO, shiftO, out);
}